// Encoder_Head_16544214024731
// MI455X (gfx1250) — compile-verified
//
#include <hip/hip_runtime.h>

// ---------------------------------------------------------------------------
// Point Cloud Transformer encoder for MI455X (gfx1250), fp32 end-to-end.
// All GEMMs use V_WMMA_F32_16X16X4_F32 (the CDNA5 fp32 matrix path).
// Feature maps are stored channel-major: arr[c*Ltot + b*Np + n].
// ---------------------------------------------------------------------------

typedef float v2f __attribute__((ext_vector_type(2)));
typedef float v8f __attribute__((ext_vector_type(8)));

// global-address-space pointers: keep GEMM loads on the global path
// (avoids flat_load fallback that ties up the LDS pipe via DScnt)
typedef const float __attribute__((address_space(1)))* gcfloat;
typedef float __attribute__((address_space(1)))*       gfloat;

#define CDIV(a, b) (((a) + (b) - 1) / (b))

// ---- sizes -----------------------------------------------------------------
#define BATCH 16
#define NPTS  4096
#define LT0   (BATCH * NPTS)       // 65536
#define S1    512
#define KNN_K 32
#define LG0   (BATCH * S1 * KNN_K) // 262144
#define S2    256
#define LG1   (BATCH * S2 * KNN_K) // 131072
#define LT1   (BATCH * S2)         // 4096

#define GEMM_NT    4               // 16-col tiles per wave (16x64 strip)
#define GEMM_WAVES 4               // waves per block

// ===========================================================================
// Generic strided (optionally batched) GEMM:  Y = A * B (+bias), fp32 WMMA.
// A(m,k) = A[b*Abat + m*Asm + k*Ask], B(k,n) = B[b*Bbat + k*Bsk + n*Bsn],
// Y(m,n) = Y[b*Obat + m*Osm + n*Osn].  Requires M%16==0, K%8==0, N%16==0.
// One wave computes a 16x64 strip: the A fragment is loaded once per k-step
// and reused across 4 WMMAs; pointers are strength-reduced and kept in
// named registers (no arrays) so they stay in addrspace(1).
// ===========================================================================
__global__ void wmma_gemm_f32(const float* __restrict__ A,
                              const float* __restrict__ B,
                              const float* __restrict__ bias,
                              float* __restrict__ Y,
                              int M, int N, int K,
                              long Asm, long Ask, long Abat,
                              long Bsk, long Bsn, long Bbat,
                              long Osm, long Osn, long Obat)
{
  __builtin_assume(K >= 8 && (K & 7) == 0);

  const int lane  = threadIdx.x & 31;
  const int wave  = threadIdx.x >> 5;
  const int strip = blockIdx.x * (blockDim.x >> 5) + wave;
  const int n0    = strip * (16 * GEMM_NT);
  if (n0 >= N) return;
  const int m0 = blockIdx.y * 16;
  const int b  = blockIdx.z;
  const int lr = lane & 15;   // 0..15
  const int hi = lane >> 4;   // 0/1 half-wave

  gcfloat Ab = (gcfloat)(A + (long)b * Abat);
  gcfloat Bb = (gcfloat)(B + (long)b * Bbat);
  gfloat  Yb = (gfloat)(Y + (long)b * Obat);

  // per-tile column bases (clamped; all call sites use N % 64 == 0)
  const int nn1 = n0 + 16, nn2 = n0 + 32, nn3 = n0 + 48;
  const bool ok1 = nn1 < N, ok2 = nn2 < N, ok3 = nn3 < N;
  const int nb1 = ok1 ? nn1 : n0;
  const int nb2 = ok2 ? nn2 : n0;
  const int nb3 = ok3 ? nn3 : n0;

  // A 16x4 f32 frag: lanes 0-15 rows m0+lr with K = k,k+1; lanes 16-31: K = k+2,k+3
  // B 4x16 f32 frag: lanes 0-15 cols nb+lr with K = k,k+1; lanes 16-31: K = k+2,k+3
  gcfloat ap  = Ab + (long)(m0 + lr) * Asm + (long)(hi * 2) * Ask;
  gcfloat bp0 = Bb + (long)(n0 + lr) * Bsn + (long)(hi * 2) * Bsk;
  gcfloat bp1 = Bb + (long)(nb1 + lr) * Bsn + (long)(hi * 2) * Bsk;
  gcfloat bp2 = Bb + (long)(nb2 + lr) * Bsn + (long)(hi * 2) * Bsk;
  gcfloat bp3 = Bb + (long)(nb3 + lr) * Bsn + (long)(hi * 2) * Bsk;

  const long aInc = 4 * Ask;
  const long bInc = 4 * Bsk;

  v8f acc0 = {0.f, 0.f, 0.f, 0.f, 0.f, 0.f, 0.f, 0.f};
  v8f acc1 = acc0, acc2 = acc0, acc3 = acc0;

#pragma unroll 2
  for (int k = 0; k < K; k += 4) {
    v2f a, b0, b1, b2, b3;
    a.x  = ap[0];      a.y  = ap[Ask];
    b0.x = bp0[0];     b0.y = bp0[Bsk];
    b1.x = bp1[0];     b1.y = bp1[Bsk];
    b2.x = bp2[0];     b2.y = bp2[Bsk];
    b3.x = bp3[0];     b3.y = bp3[Bsk];
    acc0 = __builtin_amdgcn_wmma_f32_16x16x4_f32(false, a, false, b0,
                                                 (short)0, acc0, false, false);
    acc1 = __builtin_amdgcn_wmma_f32_16x16x4_f32(false, a, false, b1,
                                                 (short)0, acc1, false, false);
    acc2 = __builtin_amdgcn_wmma_f32_16x16x4_f32(false, a, false, b2,
                                                 (short)0, acc2, false, false);
    acc3 = __builtin_amdgcn_wmma_f32_16x16x4_f32(false, a, false, b3,
                                                 (short)0, acc3, false, false);
    ap  += aInc;
    bp0 += bInc; bp1 += bInc; bp2 += bInc; bp3 += bInc;
  }

  // C/D layout: lanes 0-15 -> N=lane, M=row r; lanes 16-31 -> N=lane-16, M=8+r
  const int mb = m0 + hi * 8;
  gcfloat bsg = (gcfloat)bias;
#pragma unroll
  for (int r = 0; r < 8; ++r) {
    float bv = bias ? bsg[mb + r] : 0.f;
    long rowoff = (long)(mb + r) * Osm;
    Yb[rowoff + (long)(n0 + lr) * Osn] = acc0[r] + bv;
    if (ok1) Yb[rowoff + (long)(nb1 + lr) * Osn] = acc1[r] + bv;
    if (ok2) Yb[rowoff + (long)(nb2 + lr) * Osn] = acc2[r] + bv;
    if (ok3) Yb[rowoff + (long)(nb3 + lr) * Osn] = acc3[r] + bv;
  }
}

// ===========================================================================
// BatchNorm (train mode): per-channel mean/var over Ltot columns, then apply.
// ===========================================================================
__global__ void bn_stats(const float* __restrict__ X, float* __restrict__ mean,
                         float* __restrict__ rstd, long Ltot)
{
  const int c = blockIdx.x;
  const float* row = X + (long)c * Ltot;
  float s = 0.f, s2 = 0.f;
  for (long i = threadIdx.x; i < Ltot; i += blockDim.x) {
    float v = row[i]; s += v; s2 += v * v;
  }
  __shared__ float sh[256], sh2[256];
  sh[threadIdx.x] = s; sh2[threadIdx.x] = s2;
  __syncthreads();
  for (int st = 128; st > 0; st >>= 1) {
    if ((int)threadIdx.x < st) {
      sh[threadIdx.x]  += sh[threadIdx.x + st];
      sh2[threadIdx.x] += sh2[threadIdx.x + st];
    }
    __syncthreads();
  }
  if (threadIdx.x == 0) {
    float m   = sh[0] / (float)Ltot;
    float var = sh2[0] / (float)Ltot - m * m;
    mean[c] = m;
    rstd[c] = rsqrtf(var + 1e-5f);
  }
}

// act: 0 = none, 1 = relu
__global__ void bn_apply(const float* __restrict__ X, float* __restrict__ Y,
                         const float* __restrict__ mean, const float* __restrict__ rstd,
                         const float* __restrict__ g, const float* __restrict__ bt,
                         long Ltot, int act)
{
  const int c = blockIdx.y;
  long i = (long)blockIdx.x * blockDim.x + threadIdx.x;
  if (i >= Ltot) return;
  long idx = (long)c * Ltot + i;
  float v = (X[idx] - mean[c]) * rstd[c] * g[c] + bt[c];
  if (act == 1) v = fmaxf(v, 0.f);
  Y[idx] = v;
}

// ===========================================================================
// Input embedding conv: 3 -> 64 channels (K=3 too small for WMMA).
// ===========================================================================
__global__ void conv_in3(const float* __restrict__ x, const float* __restrict__ w,
                         float* __restrict__ y, long Ltot)
{
  __shared__ float sw[192];
  if (threadIdx.x < 192) sw[threadIdx.x] = w[threadIdx.x];
  __syncthreads();
  long l = (long)blockIdx.x * blockDim.x + threadIdx.x;
  if (l >= Ltot) return;
  float x0 = x[l * 3 + 0], x1 = x[l * 3 + 1], x2 = x[l * 3 + 2];
#pragma unroll 8
  for (int c = 0; c < 64; ++c)
    y[(long)c * Ltot + l] = sw[c * 3] * x0 + sw[c * 3 + 1] * x1 + sw[c * 3 + 2] * x2;
}

// ===========================================================================
// Farthest point sampling: one block per batch, LDS distance table.
// Matches reference: first index 0, idx[it] = far before update, argmax ties
// broken toward lowest index.
// ===========================================================================
__global__ void fps_kernel(const float* __restrict__ xyz, int N, int npoint,
                           int* __restrict__ idx_out, float* __restrict__ newxyz)
{
  extern __shared__ float dsh[];          // N floats
  __shared__ float sval[256];
  __shared__ int   sidx[256];
  const int b = blockIdx.x;
  const float* P = xyz + (long)b * N * 3;
  for (int n = threadIdx.x; n < N; n += blockDim.x) dsh[n] = 1e10f;
  __syncthreads();
  int far = 0;
  for (int it = 0; it < npoint; ++it) {
    float cx = P[3 * far + 0], cy = P[3 * far + 1], cz = P[3 * far + 2];
    if (threadIdx.x == 0) {
      idx_out[b * npoint + it] = far;
      newxyz[((long)b * npoint + it) * 3 + 0] = cx;
      newxyz[((long)b * npoint + it) * 3 + 1] = cy;
      newxyz[((long)b * npoint + it) * 3 + 2] = cz;
    }
    float best = -1.f; int bi = 0x7fffffff;
    for (int n = threadIdx.x; n < N; n += blockDim.x) {
      float dx = P[3 * n] - cx, dy = P[3 * n + 1] - cy, dz = P[3 * n + 2] - cz;
      float d  = dx * dx + dy * dy + dz * dz;
      float nd = fminf(dsh[n], d);
      dsh[n] = nd;
      if (nd > best) { best = nd; bi = n; }
    }
    sval[threadIdx.x] = best; sidx[threadIdx.x] = bi;
    __syncthreads();
    for (int st = 128; st > 0; st >>= 1) {
      if ((int)threadIdx.x < st) {
        float ov = sval[threadIdx.x + st]; int oi = sidx[threadIdx.x + st];
        if (ov > sval[threadIdx.x] ||
            (ov == sval[threadIdx.x] && oi < sidx[threadIdx.x])) {
          sval[threadIdx.x] = ov; sidx[threadIdx.x] = oi;
        }
      }
      __syncthreads();
    }
    far = sidx[0];
    __syncthreads();
  }
}

// ===========================================================================
// kNN (k smallest squared distances, ascending, ties -> lower index).
// One block per (centroid, batch); distance table in dynamic LDS.
// ===========================================================================
__global__ void knn_kernel(const float* __restrict__ xyz, const float* __restrict__ cent,
                           int N, int S, int K, int* __restrict__ idx_out)
{
  extern __shared__ float dsh[];          // N floats
  __shared__ float sval[128];
  __shared__ int   sidx[128];
  const int s = blockIdx.x, b = blockIdx.y;
  const float* P = xyz + (long)b * N * 3;
  const float* C = cent + ((long)b * S + s) * 3;
  const float cx = C[0], cy = C[1], cz = C[2];
  for (int n = threadIdx.x; n < N; n += blockDim.x) {
    float dx = P[3 * n] - cx, dy = P[3 * n + 1] - cy, dz = P[3 * n + 2] - cz;
    dsh[n] = dx * dx + dy * dy + dz * dz;
  }
  __syncthreads();
  for (int kk = 0; kk < K; ++kk) {
    float best = 3.402823e38f; int bi = 0x7fffffff;
    for (int n = threadIdx.x; n < N; n += blockDim.x) {
      float v = dsh[n];
      if (v < best) { best = v; bi = n; }
    }
    sval[threadIdx.x] = best; sidx[threadIdx.x] = bi;
    __syncthreads();
    for (int st = 64; st > 0; st >>= 1) {
      if ((int)threadIdx.x < st) {
        float ov = sval[threadIdx.x + st]; int oi = sidx[threadIdx.x + st];
        if (ov < sval[threadIdx.x] ||
            (ov == sval[threadIdx.x] && oi < sidx[threadIdx.x])) {
          sval[threadIdx.x] = ov; sidx[threadIdx.x] = oi;
        }
      }
      __syncthreads();
    }
    if (threadIdx.x == 0) {
      idx_out[((long)(b * S + s)) * K + kk] = sidx[0];
      dsh[sidx[0]] = 3.402823e38f;
    }
    __syncthreads();
  }
}

// ===========================================================================
// Group + concat([neighbor - center, center]) into channel-major (2C x B*S*K).
// ===========================================================================
__global__ void group_concat(const float* __restrict__ feat, long LtotF, int Np,
                             const int* __restrict__ fpsidx,
                             const int* __restrict__ knnidx,
                             float* __restrict__ out, int C, int S, int K, int B)
{
  long col = (long)blockIdx.x * blockDim.x + threadIdx.x;
  long Lout = (long)B * S * K;
  if (col >= Lout) return;
  int k = (int)(col % K);
  long t = col / K;
  int s = (int)(t % S);
  int b = (int)(t / S);
  int ci = fpsidx[b * S + s];
  int ni = knnidx[((long)(b * S + s)) * K + k];
  long colc = (long)b * Np + ci;
  long coln = (long)b * Np + ni;
  for (int c = 0; c < C; ++c) {
    float ctr = feat[(long)c * LtotF + colc];
    float nb  = feat[(long)c * LtotF + coln];
    out[(long)c * Lout + col]       = nb - ctr;
    out[(long)(C + c) * Lout + col] = ctr;
  }
}

// max over the K neighbor axis: (C x BS*K) -> (C x BS)
__global__ void maxpool_k(const float* __restrict__ X, float* __restrict__ Y,
                          long BS, int K)
{
  const int c = blockIdx.y;
  long col = (long)blockIdx.x * blockDim.x + threadIdx.x;
  if (col >= BS) return;
  const float* p = X + (long)c * BS * K + col * K;
  float m = -3.402823e38f;
  for (int k = 0; k < K; ++k) m = fmaxf(m, p[k]);
  Y[(long)c * BS + col] = m;
}

// ===========================================================================
// Attention helpers
// ===========================================================================
__global__ void softmax_rows256(float* __restrict__ E)
{
  __shared__ float sh[256];
  float* r = E + (long)blockIdx.x * 256;
  const int t = threadIdx.x;
  float v = r[t];
  sh[t] = v; __syncthreads();
  for (int st = 128; st > 0; st >>= 1) {
    if (t < st) sh[t] = fmaxf(sh[t], sh[t + st]);
    __syncthreads();
  }
  float mx = sh[0]; __syncthreads();
  float e = __expf(v - mx);
  sh[t] = e; __syncthreads();
  for (int st = 128; st > 0; st >>= 1) {
    if (t < st) sh[t] += sh[t + st];
    __syncthreads();
  }
  r[t] = e / sh[0];
}

__global__ void attn_colsum(const float* __restrict__ E, float* __restrict__ cs, int Np)
{
  int idx = blockIdx.x * blockDim.x + threadIdx.x;  // b*Np + m
  int b = idx / Np, m = idx % Np;
  const float* base = E + (long)b * Np * Np + m;
  float s = 0.f;
  for (int n = 0; n < Np; ++n) s += base[(long)n * Np];
  cs[idx] = s;
}

__global__ void attn_norm(float* __restrict__ E, const float* __restrict__ cs,
                          int Np, long n)
{
  long i = (long)blockIdx.x * blockDim.x + threadIdx.x;
  if (i >= n) return;
  long m = i % Np;
  long b = i / ((long)Np * Np);
  E[i] /= (1e-9f + cs[b * Np + m]);
}

// ---- elementwise ----------------------------------------------------------
__global__ void ew_sub(const float* __restrict__ A, const float* __restrict__ B,
                       float* __restrict__ Y, long n)
{ long i = (long)blockIdx.x * blockDim.x + threadIdx.x; if (i < n) Y[i] = A[i] - B[i]; }

__global__ void ew_add(const float* __restrict__ A, const float* __restrict__ B,
                       float* __restrict__ Y, long n)
{ long i = (long)blockIdx.x * blockDim.x + threadIdx.x; if (i < n) Y[i] = A[i] + B[i]; }

__global__ void ew_copy(const float* __restrict__ A, float* __restrict__ Y, long n)
{ long i = (long)blockIdx.x * blockDim.x + threadIdx.x; if (i < n) Y[i] = A[i]; }

// ===========================================================================
// Final fuse epilogue: BN apply + LeakyReLU(0.2) + max over the 256 points.
// raw is (1024 x B*256); out is (B, 1024).
// ===========================================================================
__global__ void fuse_final(const float* __restrict__ raw,
                           const float* __restrict__ mean, const float* __restrict__ rstd,
                           const float* __restrict__ g, const float* __restrict__ bt,
                           float* __restrict__ out, int B, int Np, int C)
{
  int idx = blockIdx.x * blockDim.x + threadIdx.x;   // b*C + c
  if (idx >= B * C) return;
  int b = idx / C, c = idx % C;
  const float* p = raw + (long)c * (B * Np) + (long)b * Np;
  float m = mean[c], rs = rstd[c], gg = g[c], bb = bt[c];
  float best = -3.402823e38f;
  for (int n = 0; n < Np; ++n) {
    float v = (p[n] - m) * rs * gg + bb;
    v = v > 0.f ? v : 0.2f * v;
    best = fmaxf(best, v);
  }
  out[(long)b * C + c] = best;
}

// ===========================================================================
// Host-side orchestration
// ===========================================================================
static void launch_gemm(hipStream_t st, const float* A, const float* B,
                        const float* bias, float* Y, int M, int N, int K,
                        long Asm, long Ask, long Abat,
                        long Bsk, long Bsn, long Bbat,
                        long Osm, long Osn, long Obat, int batches)
{
  dim3 block(GEMM_WAVES * 32);
  const int strips = CDIV(N, 16 * GEMM_NT);
  dim3 grid(CDIV(strips, GEMM_WAVES), M / 16, batches);
  wmma_gemm_f32<<<grid, block, 0, st>>>(A, B, bias, Y, M, N, K,
                                        Asm, Ask, Abat, Bsk, Bsn, Bbat,
                                        Osm, Osn, Obat);
}

// convenience: channel-major conv1x1  Y(Co x L) = W(Co x Ci) * X(Ci x L)
static void launch_conv(hipStream_t st, const float* W, const float* X,
                        const float* bias, float* Y, int Co, int Ci, long L)
{
  launch_gemm(st, W, X, bias, Y, Co, (int)L, Ci,
              /*A*/ Ci, 1, 0, /*B*/ L, 1, 0, /*O*/ L, 1, 0, 1);
}

static void launch_bn(hipStream_t st, float* X, float* mean, float* rstd,
                      const float* g, const float* bt, int C, long Ltot, int act)
{
  bn_stats<<<dim3(C), dim3(256), 0, st>>>(X, mean, rstd, Ltot);
  bn_apply<<<dim3((unsigned)CDIV(Ltot, 256), C), dim3(256), 0, st>>>(
      X, X, mean, rstd, g, bt, Ltot, act);
}

extern "C" void kernel_launch(void* const* d_in, const int* in_sizes, int n_in,
                              void* d_out, int out_size, void* d_ws, size_t ws_size,
                              hipStream_t stream)
{
  // ---- inputs in setup_inputs() insertion order ----
  const float* x      = (const float*)d_in[0];   // (16,4096,3)
  const float* c1_w   = (const float*)d_in[1];
  const float* bn1_g  = (const float*)d_in[2];
  const float* bn1_b  = (const float*)d_in[3];
  const float* c2_w   = (const float*)d_in[4];
  const float* bn2_g  = (const float*)d_in[5];
  const float* bn2_b  = (const float*)d_in[6];
  const float* l0_w1  = (const float*)d_in[7];
  const float* l0_g1  = (const float*)d_in[8];
  const float* l0_b1  = (const float*)d_in[9];
  const float* l0_w2  = (const float*)d_in[10];
  const float* l0_g2  = (const float*)d_in[11];
  const float* l0_b2  = (const float*)d_in[12];
  const float* l1_w1  = (const float*)d_in[13];
  const float* l1_g1  = (const float*)d_in[14];
  const float* l1_b1  = (const float*)d_in[15];
  const float* l1_w2  = (const float*)d_in[16];
  const float* l1_g2  = (const float*)d_in[17];
  const float* l1_b2  = (const float*)d_in[18];
  const float* ptc1_w = (const float*)d_in[19];
  const float* ptg1   = (const float*)d_in[20];
  const float* ptb1   = (const float*)d_in[21];
  const float* ptc2_w = (const float*)d_in[22];
  const float* ptg2   = (const float*)d_in[23];
  const float* ptb2   = (const float*)d_in[24];
  // sa[i]: qk_w, v_w, v_b, t_w, t_b, bn_g, bn_b at 25 + 7*i
  const float* fuse_w = (const float*)d_in[53];
  const float* fuse_g = (const float*)d_in[54];
  const float* fuse_b = (const float*)d_in[55];
  float* out = (float*)d_out;

  // ---- workspace layout (float offsets) ----
  float* ws = (float*)d_ws;
  const long BIG = 33554432;                 // 128ch x 262144 == 256ch x 131072
  float* BUF1 = ws;                          // big ping
  float* BUF2 = ws + BIG;                    // big pong
  long off = 2 * BIG;
  float* P_mean = ws + off; off += 1024;
  float* P_rstd = ws + off; off += 1024;
  int*   P_idx1 = (int*)(ws + off); off += BATCH * S1;            // 8192
  float* P_nx1  = ws + off; off += (long)BATCH * S1 * 3;          // 24576
  int*   P_knn1 = (int*)(ws + off); off += (long)BATCH * S1 * KNN_K;
  int*   P_idx2 = (int*)(ws + off); off += BATCH * S2;
  float* P_nx2  = ws + off; off += (long)BATCH * S2 * 3;
  int*   P_knn2 = (int*)(ws + off); off += (long)BATCH * S2 * KNN_K;
  float* P_f0   = ws + off; off += 128L * BATCH * S1;             // 1,048,576
  float* P_f1   = ws + off; off += 256L * LT1;                    // 1,048,576
  float* P_F    = ws + off; off += 1280L * LT1;                   // 5,242,880
  float* P_q    = ws + off; off += 64L * LT1;
  float* P_v    = ws + off; off += 256L * LT1;
  float* P_E    = ws + off; off += (long)BATCH * S2 * S2;
  float* P_cs   = ws + off; off += (long)BATCH * S2;
  float* P_xr   = ws + off; off += 256L * LT1;
  float* P_tmp  = ws + off; off += 256L * LT1;
  float* P_tr   = ws + off; off += 256L * LT1;

  // ===== Stage 1: embedding convs (3->64->64) with train BN + relu =====
  conv_in3<<<dim3(CDIV(LT0, 256)), dim3(256), 0, stream>>>(x, c1_w, BUF1, LT0);
  launch_bn(stream, BUF1, P_mean, P_rstd, bn1_g, bn1_b, 64, LT0, 1);
  launch_conv(stream, c2_w, BUF1, nullptr, BUF2, 64, 64, LT0);        // WMMA
  launch_bn(stream, BUF2, P_mean, P_rstd, bn2_g, bn2_b, 64, LT0, 1);  // h2 in BUF2

  // ===== Stage 2: FPS(512) + kNN(32) + group -> nf0 (128 x 262144) =====
  fps_kernel<<<dim3(BATCH), dim3(256), NPTS * sizeof(float), stream>>>(
      x, NPTS, S1, P_idx1, P_nx1);
  knn_kernel<<<dim3(S1, BATCH), dim3(128), NPTS * sizeof(float), stream>>>(
      x, P_nx1, NPTS, S1, KNN_K, P_knn1);
  group_concat<<<dim3(CDIV((long)LG0, 256)), dim3(256), 0, stream>>>(
      BUF2, LT0, NPTS, P_idx1, P_knn1, BUF1, 64, S1, KNN_K, BATCH);

  // ===== local_op l0: (128->128)x2 + BN + relu, maxpool over 32 =====
  launch_conv(stream, l0_w1, BUF1, nullptr, BUF2, 128, 128, LG0);     // WMMA
  launch_bn(stream, BUF2, P_mean, P_rstd, l0_g1, l0_b1, 128, LG0, 1);
  launch_conv(stream, l0_w2, BUF2, nullptr, BUF1, 128, 128, LG0);     // WMMA
  launch_bn(stream, BUF1, P_mean, P_rstd, l0_g2, l0_b2, 128, LG0, 1);
  maxpool_k<<<dim3(CDIV((long)BATCH * S1, 256), 128), dim3(256), 0, stream>>>(
      BUF1, P_f0, (long)BATCH * S1, KNN_K);

  // ===== Stage 3: FPS(256) + kNN(32) on the 512 centroids =====
  fps_kernel<<<dim3(BATCH), dim3(256), S1 * sizeof(float), stream>>>(
      P_nx1, S1, S2, P_idx2, P_nx2);
  knn_kernel<<<dim3(S2, BATCH), dim3(128), S1 * sizeof(float), stream>>>(
      P_nx1, P_nx2, S1, S2, KNN_K, P_knn2);
  group_concat<<<dim3(CDIV((long)LG1, 256)), dim3(256), 0, stream>>>(
      P_f0, (long)BATCH * S1, S1, P_idx2, P_knn2, BUF1, 128, S2, KNN_K, BATCH);

  // ===== local_op l1: (256->256)x2 + BN + relu, maxpool -> f1 =====
  launch_conv(stream, l1_w1, BUF1, nullptr, BUF2, 256, 256, LG1);     // WMMA
  launch_bn(stream, BUF2, P_mean, P_rstd, l1_g1, l1_b1, 256, LG1, 1);
  launch_conv(stream, l1_w2, BUF2, nullptr, BUF1, 256, 256, LG1);     // WMMA
  launch_bn(stream, BUF1, P_mean, P_rstd, l1_g2, l1_b2, 256, LG1, 1);
  maxpool_k<<<dim3(CDIV((long)LT1, 256), 256), dim3(256), 0, stream>>>(
      BUF1, P_f1, (long)LT1, KNN_K);

  // ===== pt convs: 256->256 twice, BN + relu =====
  launch_conv(stream, ptc1_w, P_f1, nullptr, BUF1, 256, 256, LT1);    // WMMA
  launch_bn(stream, BUF1, P_mean, P_rstd, ptg1, ptb1, 256, LT1, 1);
  launch_conv(stream, ptc2_w, BUF1, nullptr, BUF2, 256, 256, LT1);    // WMMA
  launch_bn(stream, BUF2, P_mean, P_rstd, ptg2, ptb2, 256, LT1, 1);   // h in BUF2

  // stash f1 into fuse-input rows 1024..1279
  ew_copy<<<dim3(CDIV(256L * LT1, 256)), dim3(256), 0, stream>>>(
      P_f1, P_F + 1024L * LT1, 256L * LT1);

  // ===== 4 offset-attention layers =====
  const float* hcur = BUF2;
  const long n256 = 256L * LT1;
  for (int i = 0; i < 4; ++i) {
    const float* qk_w = (const float*)d_in[25 + 7 * i + 0];
    const float* v_w  = (const float*)d_in[25 + 7 * i + 1];
    const float* v_b  = (const float*)d_in[25 + 7 * i + 2];
    const float* t_w  = (const float*)d_in[25 + 7 * i + 3];
    const float* t_b  = (const float*)d_in[25 + 7 * i + 4];
    const float* sg   = (const float*)d_in[25 + 7 * i + 5];
    const float* sb   = (const float*)d_in[25 + 7 * i + 6];
    float* hnext = P_F + (long)i * 256 * LT1;

    launch_conv(stream, qk_w, hcur, nullptr, P_q, 64, 256, LT1);       // WMMA
    launch_conv(stream, v_w,  hcur, v_b,     P_v, 256, 256, LT1);      // WMMA

    // energy[b] (256x256) = Q^T Q : A(m,k)=q[k*LT1 + b*256 + m]
    launch_gemm(stream, P_q, P_q, nullptr, P_E, S2, S2, 64,
                /*A*/ 1, LT1, 256, /*B*/ LT1, 1, 256,
                /*O*/ S2, 1, (long)S2 * S2, BATCH);                    // WMMA
    softmax_rows256<<<dim3(BATCH * S2), dim3(256), 0, stream>>>(P_E);
    attn_colsum<<<dim3(CDIV(BATCH * S2, 256)), dim3(256), 0, stream>>>(P_E, P_cs, S2);
    attn_norm<<<dim3(CDIV((long)BATCH * S2 * S2, 256)), dim3(256), 0, stream>>>(
        P_E, P_cs, S2, (long)BATCH * S2 * S2);

    // xr[b] (256ch x 256pts) = V[b] (256x256) * attn[b] (256x256)
    launch_gemm(stream, P_v, P_E, nullptr, P_xr, S2, S2, S2,
                /*A*/ LT1, 1, 256, /*B*/ S2, 1, (long)S2 * S2,
                /*O*/ LT1, 1, 256, BATCH);                             // WMMA

    ew_sub<<<dim3(CDIV(n256, 256)), dim3(256), 0, stream>>>(hcur, P_xr, P_tmp, n256);
    launch_conv(stream, t_w, P_tmp, t_b, P_tr, 256, 256, LT1);         // WMMA
    launch_bn(stream, P_tr, P_mean, P_rstd, sg, sb, 256, LT1, 1);
    ew_add<<<dim3(CDIV(n256, 256)), dim3(256), 0, stream>>>(hcur, P_tr, hnext, n256);
    hcur = hnext;
  }

  // ===== fuse: 1024 x 1280 GEMM + BN + leaky-relu + max over points =====
  launch_conv(stream, fuse_w, P_F, nullptr, BUF1, 1024, 1280, LT1);    // WMMA
  bn_stats<<<dim3(1024), dim3(256), 0, stream>>>(BUF1, P_mean, P_rstd, LT1);
  fuse_final<<<dim3(CDIV(BATCH * 1024, 256)), dim3(256), 0, stream>>>(
      BUF1, P_mean, P_rstd, fuse_g, fuse_b, out, BATCH, S2, 1024);

  (void)in_sizes; (void)n_in; (void)out_size; (void)ws_size;
}